// ConvNPool_60284160967032
// MI455X (gfx1250) — compile-verified
//
#include <hip/hip_runtime.h>
#include <hip/hip_bf16.h>
#include <stdint.h>

typedef __attribute__((ext_vector_type(2))) float v2f;
typedef __attribute__((ext_vector_type(8))) float v8f;

#define NGRAPH 64
#define H 64
#define DIMS 3

// ---------- helpers ----------
__device__ __forceinline__ unsigned f32_ord(float v) {
  unsigned u = __float_as_uint(v);
  return (u & 0x80000000u) ? ~u : (u | 0x80000000u);
}
__device__ __forceinline__ float ord_f32(unsigned u) {
  u = (u & 0x80000000u) ? (u & 0x7FFFFFFFu) : ~u;
  return __uint_as_float(u);
}

__global__ void fill_f32(float* __restrict__ p, float v, long n) {
  long i = (long)blockIdx.x * blockDim.x + threadIdx.x;
  if (i < n) p[i] = v;
}

// graph start offsets from sorted batch array
__global__ void graph_bounds(const int* __restrict__ batch, int N, int* __restrict__ gstart) {
  int n = blockIdx.x * blockDim.x + threadIdx.x;
  if (n >= N) return;
  int b = batch[n];
  if (n == 0) {
    for (int g = 0; g <= b; ++g) gstart[g] = 0;
  } else {
    int bp = batch[n - 1];
    if (b != bp) for (int g = bp + 1; g <= b; ++g) gstart[g] = n;
  }
  if (n == N - 1) for (int g = b + 1; g <= NGRAPH; ++g) gstart[g] = N;
}

// W' = W[:64,:] + W[64:,:]   (concat([o,o]) input collapse)
__global__ void collapse_w(const float* __restrict__ W, float* __restrict__ Wc) {
  int i = blockIdx.x * blockDim.x + threadIdx.x;
  if (i < H * H) Wc[i] = W[i] + W[H * H + i];
}

// ---------- WMMA GEMM core ----------
// Stage B (64x64 row-major) into LDS in WMMA *fragment order*: entry
// p = (k4*4 + c)*32 + lane holds the K-pair {B[kb][col], B[kb+1][col]}
// with kb = k4*4 + (lane>>4)*2, col = 16*c + (lane&15).  Fragment load is
// then a single aligned ds_load_b64 straight into an even VGPR pair.
__device__ __forceinline__ void stage_B(const float* __restrict__ B, v2f* __restrict__ sB,
                                        int tid) {
  for (int p = tid; p < 64 * 32; p += 256) {
    int lane = p & 31, c = (p >> 5) & 3, k4 = p >> 7;
    int kb = k4 * 4 + (lane >> 4) * 2;
    int col = 16 * c + (lane & 15);
    v2f t;
    t.x = B[kb * H + col];
    t.y = B[(kb + 1) * H + col];
    sB[p] = t;
  }
}

// Accumulate one 64-wide K block from A (row-major, element stride STA) into acc[4].
template <int STA>
__device__ __forceinline__ void gemm_accum(const float* __restrict__ A, int ldA,
                                           const v2f* __restrict__ Bs, long row0,
                                           int lane, int N, v8f acc[4]) {
  int half = lane >> 4, l16 = lane & 15;
  long r = row0 + l16;
  if (r >= N) r = N - 1;  // clamp tail rows (stores are guarded)
  const float* arow = A + r * (long)ldA;
#pragma unroll
  for (int k4 = 0; k4 < 16; ++k4) {
    int kb = k4 * 4 + half * 2;
    v2f a;
    if constexpr (STA == 1) {
      a = *(const v2f*)(arow + kb);            // global_load_b64
    } else {
      a.x = arow[kb * STA];
      a.y = arow[kb * STA + STA];
    }
#pragma unroll
    for (int c = 0; c < 4; ++c) {
      v2f b = Bs[(k4 * 4 + c) * 32 + lane];    // ds_load_b64, conflict-free
      acc[c] = __builtin_amdgcn_wmma_f32_16x16x4_f32(
          false, a, false, b, (short)0, acc[c], false, false);
    }
  }
}

// Y = leaky?( A0@B0 [+ A1@B1] + bias );  Y row-major [N,64].
// One wave -> 16 rows x 64 cols (4 f32 WMMA accumulators); 8 waves/block.
template <int STA0, int STA1, bool DUAL>
__global__ __launch_bounds__(256) void wmma_gemm(
    const float* __restrict__ A0, int ldA0, const float* __restrict__ B0,
    const float* __restrict__ A1, int ldA1, const float* __restrict__ B1,
    const float* __restrict__ bias, float* __restrict__ Y, int N, int leaky) {
  __shared__ v2f sB[2][64 * 32];
  int tid = threadIdx.x;
  stage_B(B0, sB[0], tid);
  if constexpr (DUAL) stage_B(B1, sB[1], tid);
  __syncthreads();

  int wave = tid >> 5;
  int lane = tid & 31;
  long row0 = (long)blockIdx.x * 128 + wave * 16;
  if (row0 >= N) return;

  v8f acc[4] = {};
  gemm_accum<STA0>(A0, ldA0, sB[0], row0, lane, N, acc);
  if constexpr (DUAL) gemm_accum<STA1>(A1, ldA1, sB[1], row0, lane, N, acc);

  int half = lane >> 4, l16 = lane & 15;
#pragma unroll
  for (int c = 0; c < 4; ++c) {
    int col = 16 * c + l16;
    float bv = bias ? bias[col] : 0.f;
#pragma unroll
    for (int v = 0; v < 8; ++v) {
      long r = row0 + v + half * 8;
      if (r < N) {
        float val = acc[c][v] + bv;
        if (leaky) val = val > 0.f ? val : 0.2f * val;
        Y[r * H + col] = val;
      }
    }
  }
}

// ---------- attention score dots: es[n]=h[n].As  ed[n]=h[n].Ad (wave per node) ----------
__global__ __launch_bounds__(256) void att_scores(
    const float* __restrict__ h, const float* __restrict__ as_, const float* __restrict__ ad_,
    float* __restrict__ es, float* __restrict__ ed, int N) {
  int n = blockIdx.x * 8 + (threadIdx.x >> 5);
  if (n >= N) return;
  int lane = threadIdx.x & 31;
  float h0 = h[(long)n * H + lane], h1 = h[(long)n * H + 32 + lane];
  float ps = h0 * as_[lane] + h1 * as_[32 + lane];
  float pd = h0 * ad_[lane] + h1 * ad_[32 + lane];
  for (int off = 16; off > 0; off >>= 1) {
    ps += __shfl_down(ps, off, 32);
    pd += __shfl_down(pd, off, 32);
  }
  if (lane == 0) { es[n] = ps; ed[n] = pd; }
}

// ---------- softmax pass 1: logits + segment max (ordered-uint atomic max) ----------
__global__ void edge_logits(const int* __restrict__ src, const int* __restrict__ dst,
                            const float* __restrict__ ea /* +d, stride DIMS */,
                            const float* __restrict__ es, const float* __restrict__ ed,
                            float* __restrict__ val, unsigned* __restrict__ mOrd,
                            int E, int EN) {
  int e = blockIdx.x * 256 + threadIdx.x;
  if (e >= EN) return;
  int s, dd; float w;
  if (e < E) { s = src[e]; dd = dst[e]; w = ea[(long)e * DIMS]; }
  else       { s = dd = e - E; w = 1.0f; }
  float v = es[s] + ed[dd];
  v = v > 0.f ? v : 0.2f * v;
  v *= w;
  val[e] = v;
  atomicMax(&mOrd[dd], f32_ord(v));
}

// ---------- softmax pass 2: exp + segment sum ----------
__global__ void edge_exp(const int* __restrict__ dst, float* __restrict__ val,
                         const unsigned* __restrict__ mOrd, float* __restrict__ ssum,
                         int E, int EN) {
  int e = blockIdx.x * 256 + threadIdx.x;
  if (e >= EN) return;
  int dd = (e < E) ? dst[e] : e - E;
  float ex = __expf(val[e] - ord_f32(mOrd[dd]));
  val[e] = ex;
  atomicAdd(&ssum[dd], ex);
}

// ---------- softmax pass 3 + attention aggregation (wave per edge) ----------
__global__ __launch_bounds__(256) void attn_aggregate(
    const int* __restrict__ src, const int* __restrict__ dst,
    const float* __restrict__ ex, const float* __restrict__ ssum,
    float* __restrict__ alpha, const float* __restrict__ h,
    float* __restrict__ out /* ld 192, pre-offset */, int ldOut, int E, int EN) {
  int e = blockIdx.x * 8 + (threadIdx.x >> 5);
  if (e >= EN) return;
  int lane = threadIdx.x & 31;
  int s, dd;
  if (e < E) { s = src[e]; dd = dst[e]; } else { s = dd = e - E; }
  float a = ex[e] / (ssum[dd] + 1e-16f);
  if (lane == 0) alpha[e] = a;
  float2 hv = ((const float2*)(h + (long)s * H))[lane];
  float* op = out + (long)dd * ldOut + 2 * lane;
  atomicAdd(op, a * hv.x);
  atomicAdd(op + 1, a * hv.y);
}

// ---------- GraphConv aggregation: agg[dst] += alpha * x[src] (wave per edge) ----------
__global__ __launch_bounds__(256) void weighted_aggregate(
    const int* __restrict__ src, const int* __restrict__ dst,
    const float* __restrict__ alpha, const float* __restrict__ xp, int ldX, int stX,
    float* __restrict__ agg, int E, int EN) {
  int e = blockIdx.x * 8 + (threadIdx.x >> 5);
  if (e >= EN) return;
  int lane = threadIdx.x & 31;
  int s, dd;
  if (e < E) { s = src[e]; dd = dst[e]; } else { s = dd = e - E; }
  float a = alpha[e];
  const float* xr = xp + (long)s * ldX;
  float v0 = xr[lane * stX];
  float v1 = xr[(lane + 32) * stX];
  float* ar = agg + (long)dd * H;
  atomicAdd(&ar[lane], a * v0);
  atomicAdd(&ar[lane + 32], a * v1);
}

// ---------- per-graph instance-norm stats (no atomics; batch sorted) ----------
__global__ __launch_bounds__(64) void inorm_stats(
    const float* __restrict__ y, const int* __restrict__ gstart,
    float* __restrict__ mean, float* __restrict__ rstd) {
  int g = blockIdx.x, c = threadIdx.x;
  int n0 = gstart[g], n1 = gstart[g + 1];
  float sum = 0.f, ss = 0.f;
  for (int n = n0; n < n1; ++n) {
    float v = y[(long)n * H + c];
    sum += v; ss += v * v;
  }
  float cnt = (float)((n1 - n0) > 0 ? (n1 - n0) : 1);
  float m = sum / cnt;
  float var = ss / cnt - m * m;
  mean[g * H + c] = m;
  rstd[g * H + c] = rsqrtf(var + 1e-5f);
}

__global__ void inorm_apply(const float* __restrict__ y, const int* __restrict__ batch,
                            const float* __restrict__ mean, const float* __restrict__ rstd,
                            float* __restrict__ out /* pre-offset */, int ldOut, int N) {
  long i = (long)blockIdx.x * 256 + threadIdx.x;
  if (i >= (long)N * H) return;
  int n = (int)(i >> 6), c = (int)(i & 63);
  int g = batch[n];
  out[(long)n * ldOut + c] = (y[i] - mean[g * H + c]) * rstd[g * H + c];
}

// ---------- host ----------
extern "C" void kernel_launch(void* const* d_in, const int* in_sizes, int n_in,
                              void* d_out, int out_size, void* d_ws, size_t ws_size,
                              hipStream_t stream) {
  (void)n_in; (void)out_size; (void)ws_size;
  const float* x   = (const float*)d_in[0];
  const int*   ei  = (const int*)d_in[1];
  const float* ea  = (const float*)d_in[2];
  const int*   bat = (const int*)d_in[3];
  const float* We  = (const float*)d_in[4];
  const float* As  = (const float*)d_in[5];
  const float* Ad  = (const float*)d_in[6];
  const float* Wr0 = (const float*)d_in[7];
  const float* Wn0 = (const float*)d_in[8];
  const float* b0  = (const float*)d_in[9];
  const float* Wr1 = (const float*)d_in[10];
  const float* Wn1 = (const float*)d_in[11];
  const float* b1  = (const float*)d_in[12];
  const float* Wr2 = (const float*)d_in[13];
  const float* Wn2 = (const float*)d_in[14];
  const float* b2  = (const float*)d_in[15];
  float* out = (float*)d_out;

  const int N  = in_sizes[3];
  const int E  = in_sizes[1] / 2;
  const int EN = E + N;                     // with self-loops
  const int LD = DIMS * H;                  // 192
  const int* src = ei;
  const int* dst = ei + E;

  // carve workspace
  char* w = (char*)d_ws;
  auto alloc = [&](size_t bytes) -> void* {
    void* p = (void*)w; w += (bytes + 255) & ~(size_t)255; return p;
  };
  float*    h     = (float*)alloc((size_t)N * H * 4);
  float*    es    = (float*)alloc((size_t)N * 4);
  float*    ed    = (float*)alloc((size_t)N * 4);
  unsigned* mOrd  = (unsigned*)alloc((size_t)N * 4);
  float*    ssum  = (float*)alloc((size_t)N * 4);
  float*    val   = (float*)alloc((size_t)EN * 4);
  float*    alpha = (float*)alloc((size_t)DIMS * EN * 4);
  float*    x1    = (float*)alloc((size_t)N * LD * 4);
  float*    agg   = (float*)alloc((size_t)N * H * 4);
  float*    y     = (float*)alloc((size_t)N * H * 4);
  float*    xcur  = (float*)alloc((size_t)N * H * 4);
  float*    mean  = (float*)alloc((size_t)NGRAPH * H * 4);
  float*    rstd  = (float*)alloc((size_t)NGRAPH * H * 4);
  float*    WnC   = (float*)alloc((size_t)H * H * 4);
  float*    WrC   = (float*)alloc((size_t)H * H * 4);
  int*      gst   = (int*)alloc((size_t)(NGRAPH + 1) * 4);

  dim3 blk(256);
  unsigned gN    = (N + 255) / 256;
  unsigned gEN   = (EN + 255) / 256;
  unsigned gWavN = (N + 7) / 8;
  unsigned gWavE = (EN + 7) / 8;
  unsigned gGemm = (N + 127) / 128;
  unsigned gNH   = ((unsigned)((long)N * H + 255)) / 256;
  unsigned gHH   = (H * H + 255) / 256;

  graph_bounds<<<gN, blk, 0, stream>>>(bat, N, gst);
  fill_f32<<<(unsigned)(((long)N * LD + 255) / 256), blk, 0, stream>>>(x1, 0.f, (long)N * LD);

  // ---- Phase 1: EGAT per dim -> x1 (block layout), alpha per dim ----
  for (int d = 0; d < DIMS; ++d) {
    // h = x[:, :, d] @ We[d]   (A stride-3 interleaved view)
    wmma_gemm<DIMS, 1, false><<<gGemm, blk, 0, stream>>>(
        x + d, LD, We + (size_t)d * H * H, nullptr, 0, nullptr, nullptr, h, N, 0);
    att_scores<<<gWavN, blk, 0, stream>>>(h, As + d * H, Ad + d * H, es, ed, N);
    fill_f32<<<gN, blk, 0, stream>>>((float*)mOrd, 0.f, N);   // ordered-uint 0 == -inf
    fill_f32<<<gN, blk, 0, stream>>>(ssum, 0.f, N);
    edge_logits<<<gEN, blk, 0, stream>>>(src, dst, ea + d, es, ed, val, mOrd, E, EN);
    edge_exp<<<gEN, blk, 0, stream>>>(dst, val, mOrd, ssum, E, EN);
    attn_aggregate<<<gWavE, blk, 0, stream>>>(src, dst, val, ssum, alpha + (size_t)d * EN,
                                              h, x1 + (size_t)d * H, LD, E, EN);
  }

  // ---- Phase 2: residual blocks per dim (input = x1 interleaved stride-3) ----
  for (int d = 0; d < DIMS; ++d) {
    const float* alp = alpha + (size_t)d * EN;

    // layer 0: 64 -> 64
    fill_f32<<<gNH, blk, 0, stream>>>(agg, 0.f, (long)N * H);
    weighted_aggregate<<<gWavE, blk, 0, stream>>>(src, dst, alp, x1 + d, LD, DIMS, agg, E, EN);
    wmma_gemm<1, DIMS, true><<<gGemm, blk, 0, stream>>>(
        agg, H, Wn0 + (size_t)d * H * H, x1 + d, LD, Wr0 + (size_t)d * H * H,
        b0 + d * H, y, N, 1);
    inorm_stats<<<NGRAPH, 64, 0, stream>>>(y, gst, mean, rstd);
    inorm_apply<<<gNH, blk, 0, stream>>>(y, bat, mean, rstd, xcur, H, N);

    // layer 1: concat([o,o]) -> collapse 128-in weights to 64-in
    collapse_w<<<gHH, blk, 0, stream>>>(Wn1 + (size_t)d * 2 * H * H, WnC);
    collapse_w<<<gHH, blk, 0, stream>>>(Wr1 + (size_t)d * 2 * H * H, WrC);
    fill_f32<<<gNH, blk, 0, stream>>>(agg, 0.f, (long)N * H);
    weighted_aggregate<<<gWavE, blk, 0, stream>>>(src, dst, alp, xcur, H, 1, agg, E, EN);
    wmma_gemm<1, 1, true><<<gGemm, blk, 0, stream>>>(
        agg, H, WnC, xcur, H, WrC, b1 + d * H, y, N, 1);
    inorm_stats<<<NGRAPH, 64, 0, stream>>>(y, gst, mean, rstd);
    inorm_apply<<<gNH, blk, 0, stream>>>(y, bat, mean, rstd, xcur, H, N);

    // layer 2: -> final output columns [64d, 64d+64)
    collapse_w<<<gHH, blk, 0, stream>>>(Wn2 + (size_t)d * 2 * H * H, WnC);
    collapse_w<<<gHH, blk, 0, stream>>>(Wr2 + (size_t)d * 2 * H * H, WrC);
    fill_f32<<<gNH, blk, 0, stream>>>(agg, 0.f, (long)N * H);
    weighted_aggregate<<<gWavE, blk, 0, stream>>>(src, dst, alp, xcur, H, 1, agg, E, EN);
    wmma_gemm<1, 1, true><<<gGemm, blk, 0, stream>>>(
        agg, H, WnC, xcur, H, WrC, b2 + d * H, y, N, 1);
    inorm_stats<<<NGRAPH, 64, 0, stream>>>(y, gst, mean, rstd);
    inorm_apply<<<gNH, blk, 0, stream>>>(y, bat, mean, rstd, out + (size_t)d * H, LD, N);
  }
}